// MQAwithKVCache_38405597561415
// MI455X (gfx1250) — compile-verified
//
#include <hip/hip_runtime.h>
#include <math.h>
#include <stdint.h>

typedef __attribute__((ext_vector_type(2))) float v2f;
typedef __attribute__((ext_vector_type(4))) float v4f;
typedef __attribute__((ext_vector_type(8))) float v8f;
typedef __attribute__((ext_vector_type(4))) int   i4v;

#define WMMA_F32(a, b, c) \
  __builtin_amdgcn_wmma_f32_16x16x4_f32(false, (a), false, (b), (short)0, (c), false, false)

// ---- CDNA5 async LDS<->global data movers (ASYNCcnt-tracked) ---------------
#if __has_builtin(__builtin_amdgcn_global_load_async_to_lds_b128) && \
    __has_builtin(__builtin_amdgcn_global_store_async_from_lds_b128) && \
    __has_builtin(__builtin_amdgcn_s_wait_asynccnt)
#define HAVE_ASYNC_COPY 1
#else
#define HAVE_ASYNC_COPY 0
#endif

#if HAVE_ASYNC_COPY
typedef __attribute__((address_space(1))) void* as1_vp;
typedef __attribute__((address_space(3))) void* as3_vp;
typedef __attribute__((address_space(1))) i4v*  as1_i4p;   // builtin param type: int4 AS(1)*
typedef __attribute__((address_space(3))) i4v*  as3_i4p;   // builtin param type: int4 AS(3)*

__device__ __forceinline__ void async_ld16(const float* g, float* l) {
  __builtin_amdgcn_global_load_async_to_lds_b128(
      (as1_i4p)(as1_vp)(void*)const_cast<float*>(g),
      (as3_i4p)(as3_vp)(void*)l, 0, 0);
}
__device__ __forceinline__ void async_st16(float* g, const float* l) {
  __builtin_amdgcn_global_store_async_from_lds_b128(
      (as1_i4p)(as1_vp)(void*)g,
      (as3_i4p)(as3_vp)(void*)const_cast<float*>(l), 0, 0);
}
#endif

// Problem constants
#define B_   8
#define QL_  8
#define CL_  4096
#define D_   2048
#define DK_  128
#define T_   4104            // CL + QL
#define NHEADS 128           // B * G * HPG
#define SCALE_ 0.08838834764831845f   // 1/sqrt(128)

#define CHUNK   256
#define NCHUNK  16           // CL / CHUNK
#define PSTRIDE 1040         // per (head,chunk) partial: m[8] + l[8] + O[8*128]

// Workspace layout (floats)
#define WS_Q    0
#define WS_KN   (64 * 2048)
#define WS_VN   (2 * 64 * 2048)
#define WS_PART (3 * 64 * 2048)
#define WS_ATTN (WS_PART + NHEADS * NCHUNK * PSTRIDE)

// ---------------------------------------------------------------------------
// y[m][n] = sum_k x[m][k] * W[n][k];  M = 64, N = 2048, K = 2048.
// One block per 16-column ntile; 4 waves split K into 512-slices, reduce in LDS.
// So every W element streams from HBM exactly once.
// ---------------------------------------------------------------------------
__global__ __launch_bounds__(128) void gemm_xWT(const float* __restrict__ x,
                                                const float* __restrict__ W,
                                                float* __restrict__ y) {
  const int lane = threadIdx.x & 31;
  const int wave = threadIdx.x >> 5;
  const int half = lane >> 4;       // 0: lanes 0-15, 1: lanes 16-31
  const int lm   = lane & 15;
  const int kofs = 2 * half;        // f32 A/B fragment: k = 4*step + 2*(lane/16) + {0,1}
  const int n0   = blockIdx.x * 16;

  __shared__ float red[4][4][256];  // [wave][mtile][8 vgpr x 32 lane]

  v8f c[4] = {};                    // 4 mtiles of 16 rows each (M = 64)
  const int k0 = wave * 512;
  for (int k = k0; k < k0 + 512; k += 4) {
    v2f b = *(const v2f*)(W + (size_t)(n0 + lm) * 2048 + k + kofs);
#pragma unroll
    for (int mt = 0; mt < 4; ++mt) {
      v2f a = *(const v2f*)(x + (size_t)(mt * 16 + lm) * 2048 + k + kofs);
      c[mt] = WMMA_F32(a, b, c[mt]);
    }
  }
#pragma unroll
  for (int mt = 0; mt < 4; ++mt)
#pragma unroll
    for (int r = 0; r < 8; ++r)
      red[wave][mt][r * 32 + lane] = c[mt][r];
  __syncthreads();
  if (wave == 0) {
#pragma unroll
    for (int mt = 0; mt < 4; ++mt)
#pragma unroll
      for (int r = 0; r < 8; ++r) {
        float s = red[0][mt][r * 32 + lane] + red[1][mt][r * 32 + lane] +
                  red[2][mt][r * 32 + lane] + red[3][mt][r * 32 + lane];
        int m = mt * 16 + r + 8 * half;   // C layout: lanes 0-15 -> M=r, 16-31 -> M=r+8
        y[(size_t)m * 2048 + n0 + lm] = s;
      }
  }
}

// ---------------------------------------------------------------------------
// Scatter freshly-projected K/V rows into the output cache tails (seq 4096..4103).
// ---------------------------------------------------------------------------
__global__ void scatter_new_kv(const float* __restrict__ kn, const float* __restrict__ vn,
                               float* __restrict__ out_key, float* __restrict__ out_val) {
  int idx = blockIdx.x * blockDim.x + threadIdx.x;
  if (idx >= 64 * 2048) return;
  int row = idx >> 11, col = idx & 2047;
  int b = row >> 3, s = row & 7;
  int g = col >> 9, h = (col >> 7) & 3, dk = col & 127;
  size_t o = ((((size_t)b * 4 + g) * 4 + h) * T_ + CL_ + s) * DK_ + dk;
  out_key[o] = kn[idx];
  out_val[o] = vn[idx];
}

// ---------------------------------------------------------------------------
// Flash-decode over one 256-row cache chunk per single-wave workgroup,
// fused with the cache -> output copy.
// Async path: HBM -> LDS (async DMA) -> {WMMA fragments via ds_load,
// LDS -> HBM copy-out via async store}; cache bytes never transit VGPRs.
// Double-buffered so tile st+1 load + tile st store overlap tile st compute.
// grid = (128 heads, 16 chunks), block = 32 (one wave).
// ---------------------------------------------------------------------------
__global__ __launch_bounds__(32) void attn_chunk(const float* __restrict__ wsQ,
                                                 const float* __restrict__ cache_k,
                                                 const float* __restrict__ cache_v,
                                                 float* __restrict__ out_key,
                                                 float* __restrict__ out_val,
                                                 float* __restrict__ part) {
  const int head  = blockIdx.x;       // b*16 + g*4 + h
  const int chunk = blockIdx.y;
  const int lane  = threadIdx.x;
  const int half  = lane >> 4;
  const int lm    = lane & 15;
  const int kofs  = 2 * half;
  const int b  = head >> 4;
  const int gh = head & 15;           // g*4 + h

  const size_t hoffC = (size_t)head * CL_ * DK_;
  const float* K = cache_k + hoffC;
  const float* V = cache_v + hoffC;
  float* OK = out_key + (size_t)head * T_ * DK_;
  float* OV = out_val + (size_t)head * T_ * DK_;

  __shared__ float pbuf[256];         // 16x16 softmax-weight tile

  // Preload Q as WMMA A-fragments: A[m][k], m = lane%16 (rows >=8 padded with 0)
  v2f qa[32];
#pragma unroll
  for (int s = 0; s < 32; ++s) {
    if (lm < QL_)
      qa[s] = *(const v2f*)(wsQ + (size_t)(b * 8 + lm) * D_ + gh * DK_ + 4 * s + kofs);
    else {
      v2f z = {0.f, 0.f};
      qa[s] = z;
    }
  }

  float mi[8], li[8];
#pragma unroll
  for (int r = 0; r < 8; ++r) { mi[r] = -INFINITY; li[r] = 0.f; }
  v8f o[8] = {};                      // O accumulator: 8 ntiles x 16x16

  const int row0c = chunk * CHUNK;

#if HAVE_ASYNC_COPY
  __shared__ __align__(16) float kbuf[2][16 * DK_];   // 16 KB
  __shared__ __align__(16) float vbuf[2][16 * DK_];   // 16 KB

  // Prologue: async-load tile 0 into buffer 0 (no VGPR data involved).
#pragma unroll
  for (int i = 0; i < 16; ++i) {
    const int e = (lane + 32 * i) * 4;
    async_ld16(K + (size_t)row0c * DK_ + e, &kbuf[0][e]);
    async_ld16(V + (size_t)row0c * DK_ + e, &vbuf[0][e]);
  }

  for (int st = 0; st < CHUNK / 16; ++st) {
    const int row0 = row0c + st * 16;
    const int cur = st & 1, nxt = cur ^ 1;

    // Wait: loads of tile st done AND stores of tile st-1 done (buffer reusable).
    __builtin_amdgcn_s_wait_asynccnt(0);
    __syncthreads();

    // Fused concat copy-out: LDS -> out caches, async, no VGPRs.
#pragma unroll
    for (int i = 0; i < 16; ++i) {
      const int e = (lane + 32 * i) * 4;
      async_st16(OK + (size_t)row0 * DK_ + e, &kbuf[cur][e]);
      async_st16(OV + (size_t)row0 * DK_ + e, &vbuf[cur][e]);
    }
    // Prefetch tile st+1 into the other buffer; overlaps the compute below.
    if (st + 1 < CHUNK / 16) {
#pragma unroll
      for (int i = 0; i < 16; ++i) {
        const int e = (lane + 32 * i) * 4;
        async_ld16(K + (size_t)(row0 + 16) * DK_ + e, &kbuf[nxt][e]);
        async_ld16(V + (size_t)(row0 + 16) * DK_ + e, &vbuf[nxt][e]);
      }
    }

    // --- S = Q * K^T : B[k][n] = K[row0+n][k] read from LDS
    v8f sc = {};
#pragma unroll
    for (int s = 0; s < 32; ++s) {
      v2f bf = *(const v2f*)&kbuf[cur][lm * DK_ + 4 * s + kofs];
      sc = WMMA_F32(qa[s], bf, sc);
    }

    // --- online softmax; row r+8*half lives across the 16 lanes of this half
    float p[8], alpha[8];
#pragma unroll
    for (int r = 0; r < 8; ++r) {
      float sv = sc[r] * SCALE_;
      float mx = sv;
      mx = fmaxf(mx, __shfl_xor(mx, 1, 32));
      mx = fmaxf(mx, __shfl_xor(mx, 2, 32));
      mx = fmaxf(mx, __shfl_xor(mx, 4, 32));
      mx = fmaxf(mx, __shfl_xor(mx, 8, 32));
      float mnew = fmaxf(mi[r], mx);
      float pv = __expf(sv - mnew);
      float rs = pv;
      rs += __shfl_xor(rs, 1, 32);
      rs += __shfl_xor(rs, 2, 32);
      rs += __shfl_xor(rs, 4, 32);
      rs += __shfl_xor(rs, 8, 32);
      float a = __expf(mi[r] - mnew);
      mi[r] = mnew;
      li[r] = li[r] * a + rs;
      alpha[r] = a;
      p[r] = pv;
    }
#pragma unroll
    for (int r = 0; r < 8; ++r) {
#pragma unroll
      for (int t = 0; t < 8; ++t) o[t][r] *= alpha[r];
      pbuf[(r + 8 * half) * 16 + lm] = p[r];
    }
    __syncthreads();

    // --- O += P * V : A = P (LDS), B[k][n] = V[row0+k][16t+n] (LDS)
#pragma unroll
    for (int s = 0; s < 4; ++s) {
      v2f pa = *(const v2f*)(&pbuf[lm * 16 + 4 * s + kofs]);
#pragma unroll
      for (int t = 0; t < 8; ++t) {
        v2f bf;
        bf.x = vbuf[cur][(4 * s + kofs) * DK_ + 16 * t + lm];
        bf.y = vbuf[cur][(4 * s + kofs + 1) * DK_ + 16 * t + lm];
        o[t] = WMMA_F32(pa, bf, o[t]);
      }
    }
    __syncthreads();
  }
  // Outstanding async stores are drained by S_ENDPGM's implicit wait-idle.
#else
  for (int st = 0; st < CHUNK / 16; ++st) {
    const int row0 = row0c + st * 16;
    if (st + 1 < CHUNK / 16)
      __builtin_prefetch(K + (size_t)(row0 + 16) * DK_ + lane * 4, 0, 1);

    const v4f* ks4 = (const v4f*)(K + (size_t)row0 * DK_);
    v4f*       kd4 = (v4f*)(OK + (size_t)row0 * DK_);
#pragma unroll
    for (int i = 0; i < 16; ++i) kd4[lane + 32 * i] = ks4[lane + 32 * i];

    v8f sc = {};
#pragma unroll
    for (int s = 0; s < 32; ++s) {
      v2f bf = *(const v2f*)(K + (size_t)(row0 + lm) * DK_ + 4 * s + kofs);
      sc = WMMA_F32(qa[s], bf, sc);
    }

    const v4f* vs4 = (const v4f*)(V + (size_t)row0 * DK_);
    v4f*       vd4 = (v4f*)(OV + (size_t)row0 * DK_);
#pragma unroll
    for (int i = 0; i < 16; ++i) vd4[lane + 32 * i] = vs4[lane + 32 * i];

    float p[8], alpha[8];
#pragma unroll
    for (int r = 0; r < 8; ++r) {
      float sv = sc[r] * SCALE_;
      float mx = sv;
      mx = fmaxf(mx, __shfl_xor(mx, 1, 32));
      mx = fmaxf(mx, __shfl_xor(mx, 2, 32));
      mx = fmaxf(mx, __shfl_xor(mx, 4, 32));
      mx = fmaxf(mx, __shfl_xor(mx, 8, 32));
      float mnew = fmaxf(mi[r], mx);
      float pv = __expf(sv - mnew);
      float rs = pv;
      rs += __shfl_xor(rs, 1, 32);
      rs += __shfl_xor(rs, 2, 32);
      rs += __shfl_xor(rs, 4, 32);
      rs += __shfl_xor(rs, 8, 32);
      float a = __expf(mi[r] - mnew);
      mi[r] = mnew;
      li[r] = li[r] * a + rs;
      alpha[r] = a;
      p[r] = pv;
    }
#pragma unroll
    for (int r = 0; r < 8; ++r) {
#pragma unroll
      for (int t = 0; t < 8; ++t) o[t][r] *= alpha[r];
      pbuf[(r + 8 * half) * 16 + lm] = p[r];
    }
    __syncthreads();
#pragma unroll
    for (int s = 0; s < 4; ++s) {
      v2f pa = *(const v2f*)(&pbuf[lm * 16 + 4 * s + kofs]);
#pragma unroll
      for (int t = 0; t < 8; ++t) {
        const float* vb = V + (size_t)(row0 + 4 * s + kofs) * DK_ + 16 * t + lm;
        v2f bf;
        bf.x = vb[0];
        bf.y = vb[DK_];
        o[t] = WMMA_F32(pa, bf, o[t]);
      }
    }
    __syncthreads();
  }
#endif

  // --- write partials: m[8], l[8], O[8 x 128] (real rows live in lanes 0-15)
  float* pp = part + (size_t)(head * NCHUNK + chunk) * PSTRIDE;
  if (lane == 0) {
    for (int r = 0; r < 8; ++r) { pp[r] = mi[r]; pp[8 + r] = li[r]; }
  }
  if (half == 0) {
#pragma unroll
    for (int t = 0; t < 8; ++t)
#pragma unroll
      for (int r = 0; r < 8; ++r)
        pp[16 + r * 128 + 16 * t + lm] = o[t][r];
  }
}

// ---------------------------------------------------------------------------
// Combine chunk partials + causal new-key block; write pre-Wo attention rows.
// grid = 128 heads, block = 128 threads.
// ---------------------------------------------------------------------------
__global__ __launch_bounds__(128) void attn_reduce(const float* __restrict__ wsQ,
                                                   const float* __restrict__ wsKn,
                                                   const float* __restrict__ wsVn,
                                                   const float* __restrict__ part,
                                                   float* __restrict__ attn_out) {
  const int head = blockIdx.x;
  const int t = threadIdx.x;
  const int b = head >> 4, gh = head & 15;

  __shared__ float sNew[8][8];
  __shared__ float Lq[8];
  __shared__ float wNew[8][8];
  __shared__ float wCh[8][NCHUNK];

  const float* Qh  = wsQ  + (size_t)(b * 8) * D_ + gh * DK_;
  const float* Knh = wsKn + (size_t)(b * 8) * D_ + gh * DK_;
  const float* Vnh = wsVn + (size_t)(b * 8) * D_ + gh * DK_;
  const float* pp  = part + (size_t)head * NCHUNK * PSTRIDE;

  if (t < 64) {
    int q = t >> 3, j = t & 7;
    float s = 0.f;
    for (int d = 0; d < DK_; ++d) s += Qh[(size_t)q * D_ + d] * Knh[(size_t)j * D_ + d];
    sNew[q][j] = (j <= q) ? s * SCALE_ : -INFINITY;   // causal: new key j visible iff j<=q
  }
  __syncthreads();
  if (t < 8) {
    int q = t;
    float M = -INFINITY;
    for (int c = 0; c < NCHUNK; ++c) M = fmaxf(M, pp[c * PSTRIDE + q]);
    for (int j = 0; j < 8; ++j) M = fmaxf(M, sNew[q][j]);
    float L = 0.f;
    for (int c = 0; c < NCHUNK; ++c) {
      float w = __expf(pp[c * PSTRIDE + q] - M);
      wCh[q][c] = w;
      L += w * pp[c * PSTRIDE + 8 + q];
    }
    for (int j = 0; j < 8; ++j) {
      float w = (sNew[q][j] == -INFINITY) ? 0.f : __expf(sNew[q][j] - M);
      wNew[q][j] = w;
      L += w;
    }
    Lq[q] = L;
  }
  __syncthreads();

  for (int q = 0; q < 8; ++q) {          // t indexes d in [0,128)
    float acc = 0.f;
    for (int c = 0; c < NCHUNK; ++c) acc += wCh[q][c] * pp[c * PSTRIDE + 16 + q * 128 + t];
    for (int j = 0; j < 8; ++j) acc += wNew[q][j] * Vnh[(size_t)j * D_ + t];
    attn_out[(size_t)(b * 8 + q) * D_ + gh * DK_ + t] = acc / Lq[q];
  }
}

// ---------------------------------------------------------------------------
extern "C" void kernel_launch(void* const* d_in, const int* in_sizes, int n_in,
                              void* d_out, int out_size, void* d_ws, size_t ws_size,
                              hipStream_t stream) {
  const float* q       = (const float*)d_in[0];
  const float* k       = (const float*)d_in[1];
  const float* v       = (const float*)d_in[2];
  const float* cache_k = (const float*)d_in[3];
  const float* cache_v = (const float*)d_in[4];
  const float* Wq      = (const float*)d_in[5];
  const float* Wk      = (const float*)d_in[6];
  const float* Wv      = (const float*)d_in[7];
  const float* Wo      = (const float*)d_in[8];

  float* out     = (float*)d_out;                         // [64, 2048]
  float* out_key = out + (size_t)64 * 2048;               // [128, 4104, 128]
  float* out_val = out_key + (size_t)NHEADS * T_ * DK_;   // [128, 4104, 128]

  float* ws    = (float*)d_ws;
  float* wsQ   = ws + WS_Q;
  float* wsKn  = ws + WS_KN;
  float* wsVn  = ws + WS_VN;
  float* partb = ws + WS_PART;
  float* attnb = ws + WS_ATTN;

  // Projections (WMMA f32)
  gemm_xWT<<<128, 128, 0, stream>>>(q, Wq, wsQ);
  gemm_xWT<<<128, 128, 0, stream>>>(k, Wk, wsKn);
  gemm_xWT<<<128, 128, 0, stream>>>(v, Wv, wsVn);

  // New K/V rows into output cache tails
  scatter_new_kv<<<(64 * 2048 + 255) / 256, 256, 0, stream>>>(wsKn, wsVn, out_key, out_val);

  // Flash-decode over cache, fused with cache->output copy (async DMA path)
  attn_chunk<<<dim3(NHEADS, NCHUNK), 32, 0, stream>>>(wsQ, cache_k, cache_v,
                                                      out_key, out_val, partb);

  // Combine partials + causal new-key block
  attn_reduce<<<NHEADS, 128, 0, stream>>>(wsQ, wsKn, wsVn, partb, attnb);

  // Output projection
  gemm_xWT<<<128, 128, 0, stream>>>(attnb, Wo, out);
}